// GRNNmodel_41102837022893
// MI455X (gfx1250) — compile-verified
//
#include <hip/hip_runtime.h>

// GRNN scan for MI455X (gfx1250, wave32).
//
// Roofline: 64 MB total traffic, ~0.3 GFLOP -> bandwidth floor ~2.7us at
// 23.3 TB/s, but the kernel is latency-bound on the T=8192 sequential chain.
// Parallelism = 512 independent batch chains. Per wave: 16 batch elements,
// the per-step batched update dx = [(A - xicov*C)*DT | xicov] @ [x; dy]
// is one exact-fp32 V_WMMA_F32_16X16X4_F32 (2x4 times 4x16 across 16 batches).
// cov/xicov are batch-independent -> computed wave-uniform in VALU.
// dy/out are staged through LDS in 64-step chunks for coalesced HBM access.

typedef float v2f __attribute__((ext_vector_type(2)));
typedef float v8f __attribute__((ext_vector_type(8)));

#define GDT     1e-3f
#define GMAXU   0.1f
#define CHUNK   64
#define SSTRIDE (CHUNK * 2 + 4)   // padded LDS row stride (floats): 132*4B=528B,
                                  // 16B-aligned, bank stride 4 -> conflict-free

__global__ __launch_bounds__(128) void grnn_scan_wmma(
    const float* __restrict__ dy,   // [B, T, 2]
    const float* __restrict__ Ain,  // [2,2]
    const float* __restrict__ Cin,  // [2,2]
    const float* __restrict__ Din,  // [2,2]
    float* __restrict__ out,        // [B, T, 2]
    int T)
{
    __shared__ float sDyAll [4][16 * SSTRIDE];
    __shared__ float sOutAll[4][16 * SSTRIDE];

    const int lane = threadIdx.x & 31;
    const int wv   = threadIdx.x >> 5;
    const int b0   = blockIdx.x * 64 + wv * 16;   // first batch of this wave

    float* sDy  = sDyAll[wv];
    float* sOut = sOutAll[wv];

    // 2x2 model matrices (wave-uniform)
    const float a00 = Ain[0], a01 = Ain[1], a10 = Ain[2], a11 = Ain[3];
    const float c00 = Cin[0], c01 = Cin[1], c10 = Cin[2], c11 = Cin[3];
    const float d00 = Din[0], d01 = Din[1], d10 = Din[2], d11 = Din[3];

    // per-lane constants for WMMA operand construction
    const int  r    = lane & 15;        // matrix row / batch column index
    const bool hi   = lane >= 16;       // K-half selector (A: K2/K3, B: dy rows)
    const bool isR0 = (r == 0);
    const bool isR1 = (r == 1);

    // state: cov is identical in every lane; x lives in lanes 0-15 (batch = lane)
    float p00 = 1.f, p01 = 0.f, p10 = 0.f, p11 = 1.f;
    float x0  = 1.f, x1  = 0.f;

    const int f4PerBatch = (CHUNK * 2) / 4;       // 32 float4 per batch per chunk
    const int f4Total    = 16 * f4PerBatch;       // 512 float4 per wave per chunk

    for (int t0 = 0; t0 < T; t0 += CHUNK) {
        // ---- stage dy chunk: coalesced global b128 -> LDS ----
        for (int i = lane; i < f4Total; i += 32) {
            const int bl = i / f4PerBatch;
            const int j  = i % f4PerBatch;
            const float4 v =
                ((const float4*)(dy + (size_t)(b0 + bl) * T * 2 + t0 * 2))[j];
            ((float4*)(sDy + bl * SSTRIDE))[j] = v;
        }
        // prefetch next chunk (global_prefetch_b8): 4 x 128B lines per batch
        if (t0 + CHUNK < T) {
            for (int i = lane; i < 64; i += 32) {
                const int bl = i >> 2, q = i & 3;
                __builtin_prefetch(
                    dy + (size_t)(b0 + bl) * T * 2 + (t0 + CHUNK) * 2 + q * 32,
                    0, 0);
            }
        }
        __syncthreads();

        // ---- sequential scan over the chunk ----
        for (int tt = 0; tt < CHUNK; ++tt) {
            // xicov = cov @ C^T + D^T   (wave-uniform)
            const float xi00 = p00 * c00 + p01 * c01 + d00;
            const float xi01 = p00 * c10 + p01 * c11 + d10;
            const float xi10 = p10 * c00 + p11 * c01 + d01;
            const float xi11 = p10 * c10 + p11 * c11 + d11;
            // M = (A - xicov @ C) * DT
            const float m00 = (a00 - (xi00 * c00 + xi01 * c10)) * GDT;
            const float m01 = (a01 - (xi00 * c01 + xi01 * c11)) * GDT;
            const float m10 = (a10 - (xi10 * c00 + xi11 * c10)) * GDT;
            const float m11 = (a11 - (xi10 * c01 + xi11 * c11)) * GDT;

            // output uses pre-update state: out = (C @ x) * DT  (lanes 0-15)
            const float o0 = (c00 * x0 + c01 * x1) * GDT;
            const float o1 = (c10 * x0 + c11 * x1) * GDT;

            // dy for batch n = r (lanes 16-31 feed it into the B operand)
            const v2f dv = *(const v2f*)(sDy + r * SSTRIDE + tt * 2);

            // A operand (16x4 f32, 2 VGPRs): lanes 0-15 row r -> K0,K1 = M row;
            // lanes 16-31 row r -> K2,K3 = xicov row; rows >= 2 are zero.
            const float a0lo = isR0 ? m00 : (isR1 ? m10 : 0.f);
            const float a1lo = isR0 ? m01 : (isR1 ? m11 : 0.f);
            const float a0hi = isR0 ? xi00 : (isR1 ? xi10 : 0.f);
            const float a1hi = isR0 ? xi01 : (isR1 ? xi11 : 0.f);
            v2f Aop; Aop[0] = hi ? a0hi : a0lo;
                     Aop[1] = hi ? a1hi : a1lo;

            // B operand (4x16 f32, 2 VGPRs): VGPR0 = rows K0(x0)/K2(dy0),
            // VGPR1 = rows K1(x1)/K3(dy1); column n = batch.
            v2f Bop; Bop[0] = hi ? dv[0] : x0;
                     Bop[1] = hi ? dv[1] : x1;

            v8f acc = {0.f, 0.f, 0.f, 0.f, 0.f, 0.f, 0.f, 0.f};
            const v8f dres = __builtin_amdgcn_wmma_f32_16x16x4_f32(
                false, Aop, false, Bop, (short)0, acc, false, false);

            // D rows 0,1 land in VGPRs 0,1, lanes 0-15 = batch n
            float dx0 = dres[0];
            float dx1 = dres[1];
            dx0 = fminf(fmaxf(dx0, -GMAXU), GMAXU);
            dx1 = fminf(fmaxf(dx1, -GMAXU), GMAXU);
            x0 += dx0;
            x1 += dx1;

            // cov_dt = A@cov + cov@A^T + D - xicov@xicov^T  (wave-uniform)
            const float q00 = (a00 * p00 + a01 * p10) + (p00 * a00 + p01 * a01)
                              + d00 - (xi00 * xi00 + xi01 * xi01);
            const float q01 = (a00 * p01 + a01 * p11) + (p00 * a10 + p01 * a11)
                              + d01 - (xi00 * xi10 + xi01 * xi11);
            const float q10 = (a10 * p00 + a11 * p10) + (p10 * a00 + p11 * a01)
                              + d10 - (xi10 * xi00 + xi11 * xi01);
            const float q11 = (a10 * p01 + a11 * p11) + (p10 * a10 + p11 * a11)
                              + d11 - (xi10 * xi10 + xi11 * xi11);
            p00 = fminf(fmaxf(p00 + q00 * GDT, -1.f), 1.f);
            p01 = fminf(fmaxf(p01 + q01 * GDT, -1.f), 1.f);
            p10 = fminf(fmaxf(p10 + q10 * GDT, -1.f), 1.f);
            p11 = fminf(fmaxf(p11 + q11 * GDT, -1.f), 1.f);

            // stash output in LDS (lanes 0-15 only; divergence is fine here,
            // EXEC is back to all-ones before the next WMMA)
            if (!hi) {
                v2f ov; ov[0] = o0; ov[1] = o1;
                *(v2f*)(sOut + r * SSTRIDE + tt * 2) = ov;
            }
        }
        __syncthreads();

        // ---- flush out chunk: LDS -> coalesced global b128 ----
        for (int i = lane; i < f4Total; i += 32) {
            const int bl = i / f4PerBatch;
            const int j  = i % f4PerBatch;
            const float4 v = ((const float4*)(sOut + bl * SSTRIDE))[j];
            ((float4*)(out + (size_t)(b0 + bl) * T * 2 + t0 * 2))[j] = v;
        }
        __syncthreads();
    }
}

extern "C" void kernel_launch(void* const* d_in, const int* in_sizes, int n_in,
                              void* d_out, int out_size, void* d_ws, size_t ws_size,
                              hipStream_t stream)
{
    const float* dy = (const float*)d_in[0];   // inputs [B, T, 2]
    const float* A  = (const float*)d_in[1];   // [2,2]
    const float* C  = (const float*)d_in[2];   // [2,2]
    const float* D  = (const float*)d_in[3];   // [2,2]
    float* out = (float*)d_out;

    const int B = 512;
    const int T = in_sizes[0] / (B * 2);       // 8192

    dim3 grid(B / 64);       // 8 blocks
    dim3 block(128);         // 4 wave32s, each owns 16 batch chains
    hipLaunchKernelGGL(grnn_scan_wmma, grid, block, 0, stream,
                       dy, A, C, D, out, T);
}